// GraphSageLayer_16415365006092
// MI455X (gfx1250) — compile-verified
//
#include <hip/hip_runtime.h>
#include <hip/hip_bf16.h>

#define DIN   96
#define DK    192        // 2*DIN
#define DOUT  96
#define NJT   6          // 96/16 column tiles
#define KST   48         // 192/4 k-steps
#define KST_A 24         // k-steps that read agg (96/4)

typedef __attribute__((ext_vector_type(2))) float v2f;
typedef __attribute__((ext_vector_type(8))) float v8f;

// ---------------- kernel 1: zero the agg workspace ----------------
__global__ __launch_bounds__(256) void sage_zero_kernel(float* __restrict__ p, int n4) {
    int i = blockIdx.x * blockDim.x + threadIdx.x;
    if (i < n4) {
        float4 z; z.x = 0.f; z.y = 0.f; z.z = 0.f; z.w = 0.f;
        reinterpret_cast<float4*>(p)[i] = z;
    }
}

// ---------------- kernel 2: edge scatter-sum (one wave32 per edge) ----------------
__global__ __launch_bounds__(256) void sage_scatter_kernel(const float* __restrict__ feat,
                                                           const int* __restrict__ edge,
                                                           float* __restrict__ agg, int E) {
    int gid  = blockIdx.x * blockDim.x + threadIdx.x;
    int e    = gid >> 5;
    int lane = gid & 31;
    if (e >= E) return;
    int src = edge[2 * e + 0];
    int dst = edge[2 * e + 1];
    const float* fs = feat + (size_t)src * DIN;
    float*       ad = agg  + (size_t)dst * DIN;
    #pragma unroll
    for (int d = lane; d < DIN; d += 32)
        atomicAdd(&ad[d], fs[d]);
}

// ---------------- kernel 3: fused GEMM + ReLU + L2-normalize ----------------
// 8 waves/block; each wave computes a 16-row x 96-col output tile with
// v_wmma_f32_16x16x4_f32 accumulation over K=192.
__global__ __launch_bounds__(256)
void sage_gemm_kernel(const float* __restrict__ feat,
                      const int*   __restrict__ degree,
                      const float* __restrict__ W,
                      const float* __restrict__ agg,
                      float* __restrict__ out, int numTiles) {
    // Pre-packed per-lane B fragments: [j][kstep][lane] of v2f -> 73728 bytes.
    __shared__ v2f Bpack[NJT * KST * 32];

    const int tid = threadIdx.x;
    // Cooperatively pack W (B = W^T) into WMMA B-fragment layout.
    for (int p = tid; p < NJT * KST * 32; p += 256) {
        int j    = p / (KST * 32);
        int rem  = p - j * (KST * 32);
        int ks   = rem >> 5;
        int ln   = rem & 31;
        int n    = j * 16 + (ln & 15);
        int kb   = ks * 4 + ((ln >> 4) << 1);   // lanes 16-31 hold K=kk+2..kk+3
        v2f b;
        b.x = W[n * DK + kb];
        b.y = W[n * DK + kb + 1];
        Bpack[p] = b;
    }
    __syncthreads();

    const int lane = tid & 31;
    const int wave = tid >> 5;
    const int tile = blockIdx.x * 8 + wave;
    if (tile >= numTiles) return;            // wave-uniform: EXEC stays all-ones for WMMA

    const int m      = lane & 15;            // A row handled by this lane
    const int kphase = (lane >> 4) << 1;     // 0 for lanes 0-15, 2 for lanes 16-31
    const int row    = tile * 16 + m;

    int   deg = degree[row];
    float inv = 1.0f / (float)(deg < 1 ? 1 : deg);
    const float* aggRow  = agg  + (size_t)row * DIN;
    const float* featRow = feat + (size_t)row * DIN;

    v8f acc[NJT];
    #pragma unroll
    for (int j = 0; j < NJT; ++j) acc[j] = (v8f){};

    // K-steps 0..23: columns from agg * inv   (never crosses the 96 boundary)
    #pragma unroll 4
    for (int ks = 0; ks < KST_A; ++ks) {
        int k0 = ks * 4 + kphase;
        v2f a;
        a.x = aggRow[k0]     * inv;
        a.y = aggRow[k0 + 1] * inv;
        #pragma unroll
        for (int j = 0; j < NJT; ++j) {
            v2f b = Bpack[(j * KST + ks) * 32 + lane];
            acc[j] = __builtin_amdgcn_wmma_f32_16x16x4_f32(
                false, a, false, b, (short)0, acc[j], false, false);
        }
    }
    // K-steps 24..47: columns from feat
    #pragma unroll 4
    for (int ks = KST_A; ks < KST; ++ks) {
        int k0 = ks * 4 + kphase - DIN;
        v2f a;
        a.x = featRow[k0];
        a.y = featRow[k0 + 1];
        #pragma unroll
        for (int j = 0; j < NJT; ++j) {
            v2f b = Bpack[(j * KST + ks) * 32 + lane];
            acc[j] = __builtin_amdgcn_wmma_f32_16x16x4_f32(
                false, a, false, b, (short)0, acc[j], false, false);
        }
    }

    // ReLU + per-row sum of squares (C/D layout: VGPR r holds row r (lanes 0-15)
    // and row r+8 (lanes 16-31); column = j*16 + (lane&15)).
    float ss[8];
    #pragma unroll
    for (int r = 0; r < 8; ++r) {
        float s = 0.0f;
        #pragma unroll
        for (int j = 0; j < NJT; ++j) {
            float v = fmaxf(acc[j][r], 0.0f);
            acc[j][r] = v;
            s += v * v;
        }
        ss[r] = s;
    }
    // Reduce across the 16-lane half-wave, then scale by 1/max(||row||, 1e-12).
    #pragma unroll
    for (int r = 0; r < 8; ++r) {
        float s = ss[r];
        s += __shfl_xor(s, 1, 16);
        s += __shfl_xor(s, 2, 16);
        s += __shfl_xor(s, 4, 16);
        s += __shfl_xor(s, 8, 16);
        float scale = 1.0f / fmaxf(sqrtf(s), 1e-12f);
        #pragma unroll
        for (int j = 0; j < NJT; ++j) acc[j][r] *= scale;
    }

    const int rowBase = tile * 16 + ((lane >> 4) << 3);  // +8 for upper half-wave
    const int col     = lane & 15;
    #pragma unroll
    for (int r = 0; r < 8; ++r) {
        float* orow = out + (size_t)(rowBase + r) * DOUT;
        #pragma unroll
        for (int j = 0; j < NJT; ++j)
            orow[j * 16 + col] = acc[j][r];
    }
}

extern "C" void kernel_launch(void* const* d_in, const int* in_sizes, int n_in,
                              void* d_out, int out_size, void* d_ws, size_t ws_size,
                              hipStream_t stream) {
    const float* feat   = (const float*)d_in[0];
    const int*   edge   = (const int*)  d_in[1];
    const int*   degree = (const int*)  d_in[2];
    const float* W      = (const float*)d_in[3];
    float*       out    = (float*)d_out;

    const int N = in_sizes[2];
    const int E = in_sizes[1] / 2;
    float* agg = (float*)d_ws;                 // N * DIN floats (19.2 MB)

    // 1) zero agg
    int n4 = (N * DIN) / 4;
    sage_zero_kernel<<<(n4 + 255) / 256, 256, 0, stream>>>(agg, n4);

    // 2) scatter-sum over edges (wave per edge)
    long long sthreads = (long long)E * 32;
    int sblocks = (int)((sthreads + 255) / 256);
    sage_scatter_kernel<<<sblocks, 256, 0, stream>>>(feat, edge, agg, E);

    // 3) fused GEMM + ReLU + normalize
    int numTiles = (N + 15) / 16;              // 3125
    int gblocks  = (numTiles + 7) / 8;
    sage_gemm_kernel<<<gblocks, 256, 0, stream>>>(feat, degree, W, agg, out, numTiles);
}